// ContrastiveLoss_V1_80410377716006
// MI455X (gfx1250) — compile-verified
//
#include <hip/hip_runtime.h>
#include <stdint.h>

#define C_EPS   1e-6f
#define N_EMB   8192
#define D_EMB   256
#define TILE    64
#define KC      64
#define LDA     68   // padded LDS row stride (floats); 272B = 16B-aligned rows

typedef __attribute__((ext_vector_type(2))) float v2f;
typedef __attribute__((ext_vector_type(8))) float v8f;

__global__ void zero_out_kernel(float* out) { out[0] = 0.0f; }

// One block (256 threads = 8 waves) per row: normalize, emit e, s=sum(e^2), r=sum(e).
__global__ __launch_bounds__(256) void normalize_kernel(const float* __restrict__ emb,
                                                        float* __restrict__ e,
                                                        float* __restrict__ s,
                                                        float* __restrict__ r) {
  const int row  = blockIdx.x;
  const int tid  = threadIdx.x;
  const int lane = tid & 31;
  const int wid  = tid >> 5;

  float x  = emb[(size_t)row * D_EMB + tid];
  float ss = x * x;
  float sx = x;
  #pragma unroll
  for (int off = 16; off > 0; off >>= 1) {
    ss += __shfl_xor(ss, off, 32);
    sx += __shfl_xor(sx, off, 32);
  }
  __shared__ float sh_ss[8], sh_sx[8];
  if (lane == 0) { sh_ss[wid] = ss; sh_sx[wid] = sx; }
  __syncthreads();
  float tss = 0.0f, tsx = 0.0f;
  #pragma unroll
  for (int w = 0; w < 8; ++w) { tss += sh_ss[w]; tsx += sh_sx[w]; }

  const float norm = sqrtf(tss);
  const float inv  = 1.0f / fmaxf(norm, C_EPS);
  e[(size_t)row * D_EMB + tid] = x * inv;
  if (tid == 0) {
    s[row] = tss * inv * inv;   // sum of normalized squares
    r[row] = tsx * inv;         // sum of normalized values
  }
}

// 64x64 G-tile per block, fused contrastive-loss epilogue, lower-triangular blocks only.
// Global->LDS staging uses CDNA5 async DMA (global_load_async_to_lds_b128, ASYNCcnt).
__global__ __launch_bounds__(256) void loss_kernel(const float* __restrict__ e,
                                                   const float* __restrict__ s,
                                                   const float* __restrict__ r,
                                                   const int*   __restrict__ lab,
                                                   float* __restrict__ out,
                                                   float inv_ncomp) {
  const int bj = blockIdx.x;   // column block
  const int bi = blockIdx.y;   // row block
  if (bi < bj) return;         // strict upper-triangular blocks contribute nothing

  const int row_base = bi * TILE;
  const int col_base = bj * TILE;

  __shared__ float As[TILE][LDA];
  __shared__ float Bs[TILE][LDA];
  __shared__ float s_row[TILE], r_row[TILE], s_col[TILE], r_col[TILE];
  __shared__ int   l_row[TILE], l_col[TILE];
  __shared__ float sh_sum[8];

  const int t = threadIdx.x;
  if (t < TILE) {
    s_row[t] = s[row_base + t];
    r_row[t] = r[row_base + t];
    l_row[t] = lab[row_base + t];
  } else if (t < 2 * TILE) {
    const int u = t - TILE;
    s_col[u] = s[col_base + u];
    r_col[u] = r[col_base + u];
    l_col[u] = lab[col_base + u];
  }

  const int lane = t & 31;
  const int w    = t >> 5;
  const int mi   = w & 3;              // M-subtile (0..3)
  const int ni0  = (w >> 2) * 2;       // first of two N-subtiles
  const int lh   = lane >> 4;          // 0: K={0,1}, 1: K={2,3} (ISA 32-bit A/B striping)
  const int ll   = lane & 15;

  v8f acc0 = {0.f, 0.f, 0.f, 0.f, 0.f, 0.f, 0.f, 0.f};
  v8f acc1 = {0.f, 0.f, 0.f, 0.f, 0.f, 0.f, 0.f, 0.f};

  for (int kc = 0; kc < D_EMB; kc += KC) {
    __syncthreads();  // protect LDS (and first-iter s/r/l stores) before rewrite
    // Async DMA: fill 64x64 K-chunk of both panels directly into LDS (no VGPR staging).
    #pragma unroll
    for (int it = 0; it < 4; ++it) {
      const int idx = t + it * 256;    // 0..1023 float4 slots
      const int rr  = idx >> 4;        // row within tile
      const int kq  = idx & 15;        // float4 index within 64-float row-chunk
      const unsigned lds_a = (unsigned)(uintptr_t)&As[rr][kq * 4];
      const unsigned lds_b = (unsigned)(uintptr_t)&Bs[rr][kq * 4];
      const unsigned long long ga =
          (unsigned long long)(uintptr_t)(e + (size_t)(row_base + rr) * D_EMB + kc + kq * 4);
      const unsigned long long gb =
          (unsigned long long)(uintptr_t)(e + (size_t)(col_base + rr) * D_EMB + kc + kq * 4);
      asm volatile("global_load_async_to_lds_b128 %0, %1, off"
                   :: "v"(lds_a), "v"(ga) : "memory");
      asm volatile("global_load_async_to_lds_b128 %0, %1, off"
                   :: "v"(lds_b), "v"(gb) : "memory");
    }
    asm volatile("s_wait_asynccnt 0x0" ::: "memory");
    __syncthreads();

    #pragma unroll
    for (int k = 0; k < KC; k += 4) {
      const int kk = k + 2 * lh;
      const v2f a  = *(const v2f*)&As[mi * 16 + ll][kk];
      const v2f b0 = *(const v2f*)&Bs[ni0 * 16 + ll][kk];
      const v2f b1 = *(const v2f*)&Bs[(ni0 + 1) * 16 + ll][kk];
      acc0 = __builtin_amdgcn_wmma_f32_16x16x4_f32(false, a, false, b0, (short)0, acc0, false, false);
      acc1 = __builtin_amdgcn_wmma_f32_16x16x4_f32(false, a, false, b1, (short)0, acc1, false, false);
    }
  }

  // Fused epilogue: per-lane loss over the 2x8 D elements this lane owns.
  float lsum = 0.0f;
  #pragma unroll
  for (int v = 0; v < 8; ++v) {
    const int M  = v + 8 * lh;         // C/D layout: VGPR v -> M=v (lanes 0-15) / v+8 (16-31)
    const int mr = mi * 16 + M;        // row within 64-tile
    const int i  = row_base + mr;
    const float si = s_row[mr];
    const float ri = r_row[mr];
    const int   li = l_row[mr];
    #pragma unroll
    for (int q = 0; q < 2; ++q) {
      const int ncl = (ni0 + q) * 16 + ll;   // col within 64-tile
      const int j   = col_base + ncl;
      if (i > j) {
        const float g     = (q == 0) ? acc0[v] : acc1[v];
        const float dist2 = si + s_col[ncl] - 2.0f * g
                          + 2.0f * C_EPS * (ri - r_col[ncl])
                          + (float)D_EMB * C_EPS * C_EPS;
        const float dist  = sqrtf(fmaxf(dist2, 1e-12f));
        const int   lj    = l_col[ncl];
        float loss;
        if (li == lj) {
          const float dp = fmaxf(dist - 0.1f, 0.0f);
          loss = 0.5f * dp * dp;
        } else {
          const float boost = ((i == j + 1) && (li == 1)) ? 2.0f : 1.0f;
          const float dn = fmaxf(1.0f - dist, 0.0f);
          loss = 0.5f * boost * dn * dn;
        }
        lsum += loss;
      }
    }
  }

  // Block reduction -> single scaled atomicAdd into d_out.
  #pragma unroll
  for (int off = 16; off > 0; off >>= 1) lsum += __shfl_xor(lsum, off, 32);
  if (lane == 0) sh_sum[w] = lsum;
  __syncthreads();
  if (t == 0) {
    float tot = 0.0f;
    #pragma unroll
    for (int q = 0; q < 8; ++q) tot += sh_sum[q];
    atomicAdd(out, tot * inv_ncomp);
  }
}

extern "C" void kernel_launch(void* const* d_in, const int* in_sizes, int n_in,
                              void* d_out, int out_size, void* d_ws, size_t ws_size,
                              hipStream_t stream) {
  const float* emb = (const float*)d_in[0];
  const int*   lab = (const int*)d_in[1];
  float*       out = (float*)d_out;

  float* e = (float*)d_ws;                       // 8192*256 f32 = 8 MB
  float* s = e + (size_t)N_EMB * D_EMB;          // 8192 f32
  float* r = s + N_EMB;                          // 8192 f32

  zero_out_kernel<<<1, 1, 0, stream>>>(out);
  normalize_kernel<<<N_EMB, 256, 0, stream>>>(emb, e, s, r);

  const double ncomp = (double)N_EMB * (double)(N_EMB - 1) / 2.0;
  dim3 grid(N_EMB / TILE, N_EMB / TILE);
  loss_kernel<<<grid, 256, 0, stream>>>(e, s, r, lab, out, (float)(1.0 / ncomp));
}